// MAMBA_CHUNK_SCAN_kernel_78254304133718
// MI455X (gfx1250) — compile-verified
//
#include <hip/hip_runtime.h>

typedef __attribute__((ext_vector_type(2))) float v2f;
typedef __attribute__((ext_vector_type(8))) float v8f;

namespace {
constexpr int kB  = 2;
constexpr int kS  = 4096;
constexpr int kCS = 256;
constexpr int kH  = 32;
constexpr int kP  = 64;
constexpr int kN  = 128;
constexpr int kNC = 16;

constexpr int kPad = 34; // 32 K cols + 2 (even -> 8B-aligned b64 frag loads, conflict-free)
} // namespace

// Async global->LDS copy of 16 bytes per lane (gfx1250 GLOBAL_LOAD_ASYNC_TO_LDS_B128,
// GVS addressing: saddr base + 32-bit voffset). Tracked on ASYNCcnt.
__device__ __forceinline__ void async_copy_b128(uint32_t lds_byte_off,
                                                uint32_t glb_byte_off,
                                                const float* base) {
  asm volatile("global_load_async_to_lds_b128 %0, %1, %2 offset:0"
               :
               : "v"(lds_byte_off), "v"(glb_byte_off), "s"(base)
               : "memory");
}
__device__ __forceinline__ void wait_async0() {
  asm volatile("s_wait_asynccnt 0x0" ::: "memory");
}

__global__ __launch_bounds__(256)
void mamba_chunk_scan_wmma_f32(
    const float* __restrict__ cb, const float* __restrict__ x,
    const float* __restrict__ dt, const float* __restrict__ dA,
    const float* __restrict__ Cm, const float* __restrict__ ps,
    const float* __restrict__ Dv, float* __restrict__ out) {
  __shared__ float ldsA[kCS * kPad]; // A K-panel: [m][k] (256 x 32)
  __shared__ float ldsBt[kP * kPad]; // B K-panel, N-major: [p][k] (64 x 32)
  __shared__ float sExp[kCS];        // exp(dA_m): final row scale
  __shared__ float sFk[kCS];         // exp(-dA_k)*dt_k: folded into A of matmul2

  const int tid  = threadIdx.x;
  const int lane = tid & 31;
  const int wave = tid >> 5;

  const int gid = blockIdx.x;
  const int h = gid & (kH - 1);
  const int c = (gid >> 5) & (kNC - 1);
  const int b = gid >> 9;

  // LDS byte offsets for async-to-LDS (low 32 bits of flat shared address == LDS offset)
  const uint32_t ldsA_base  = (uint32_t)(uintptr_t)(void*)&ldsA[0];
  const uint32_t ldsBt_base = (uint32_t)(uintptr_t)(void*)&ldsBt[0];

  // ---- per-row scale precompute ----
  {
    const int base = ((b * kH + h) * kNC + c) * kCS;
    const float a = dA[base + tid];
    sExp[tid] = __expf(a);
    sFk[tid]  = __expf(-a) * dt[base + tid];
  }

  const v8f vzero = {0.f, 0.f, 0.f, 0.f, 0.f, 0.f, 0.f, 0.f};
  v8f acc[2][4];
#pragma unroll
  for (int mt = 0; mt < 2; ++mt)
#pragma unroll
    for (int nt = 0; nt < 4; ++nt) acc[mt][nt] = vzero;

  const int m0w  = wave * 32;            // this wave's 32-row strip
  const int l16  = lane & 15;
  const int koff = (lane < 16) ? 0 : 2;  // f32 WMMA K split across half-waves

  // WMMA over one staged 32-K panel: 8 k-steps x (2 M-tiles x 4 N-tiles)
  auto mma_panel = [&]() {
#pragma unroll
    for (int kc = 0; kc < 32; kc += 4) {
      v2f afr[2];
#pragma unroll
      for (int mt = 0; mt < 2; ++mt)
        afr[mt] = *(const v2f*)&ldsA[(m0w + mt * 16 + l16) * kPad + kc + koff];
#pragma unroll
      for (int nt = 0; nt < 4; ++nt) {
        const v2f bfr = *(const v2f*)&ldsBt[(nt * 16 + l16) * kPad + kc + koff];
#pragma unroll
        for (int mt = 0; mt < 2; ++mt)
          acc[mt][nt] = __builtin_amdgcn_wmma_f32_16x16x4_f32(
              false, afr[mt], false, bfr, (short)0, acc[mt][nt], false, false);
      }
    }
  };

  // ---- Phase 1: inter-chunk  acc += C_h[256x128] @ prev_states^T[128x64] ----
  // Both panels are pure row copies in the N-major B layout -> async-to-LDS.
  const float* Cbase  = Cm + (size_t)(b * kS + c * kCS) * kN;
  const float* psBase = ps + (size_t)(((b * kNC + c) * kH + h) * kP) * kN;

  for (int n0 = 0; n0 < kN; n0 += 32) {
    __syncthreads(); // prior panel fully consumed before overwrite
    // A: ldsA[row][0:32] = C[row][n0:n0+32]   (256 rows x 128B, 8 b128/thread)
#pragma unroll
    for (int i = 0; i < 8; ++i) {
      const int idx = i * 256 + tid;
      const int row = idx >> 3;
      const int seg = idx & 7; // 4 floats per b128
      async_copy_b128(ldsA_base + (uint32_t)(row * kPad + seg * 4) * 4u,
                      (uint32_t)(row * kN + n0 + seg * 4) * 4u, Cbase);
    }
    // B: ldsBt[p][0:32] = ps[p][n0:n0+32]     (64 rows x 128B, 2 b128/thread)
#pragma unroll
    for (int i = 0; i < 2; ++i) {
      const int idx = i * 256 + tid;
      const int p   = idx >> 3;
      const int seg = idx & 7;
      async_copy_b128(ldsBt_base + (uint32_t)(p * kPad + seg * 4) * 4u,
                      (uint32_t)(p * kN + n0 + seg * 4) * 4u, psBase);
    }
    wait_async0();
    __syncthreads();
    mma_panel();
  }

  // ---- Phase 2: intra-chunk  acc += (mask(m>=k)*cb[m,k]*fk[k]) @ x_c[256x64] ----
  const float* cbBase = cb + (size_t)(b * kNC + c) * kCS * kCS;
  const float* xBase  = x + (size_t)((b * kS + c * kCS) * kH + h) * kP;
  constexpr int xRow = kH * kP; // 2048 floats between seq positions

  for (int k0 = 0; k0 < kCS; k0 += 32) {
    __syncthreads();
    // A: masked + fk-scaled cb panel (transforms data -> manual staging)
#pragma unroll 4
    for (int i = 0; i < 32; ++i) {
      const int idx = i * 256 + tid;
      const int m  = idx >> 5;
      const int kk = idx & 31;
      const int k  = k0 + kk;
      float v = 0.0f;
      if (m >= k) v = cbBase[m * kCS + k] * sFk[k];
      ldsA[m * kPad + kk] = v;
    }
    // B: ldsBt[p][k] = x[k0+k][p]  (coalesced row reads, conflict-free strided writes)
#pragma unroll
    for (int i = 0; i < 8; ++i) {
      const int e = i * 256 + tid;
      const int k = e >> 6;
      const int p = e & 63;
      ldsBt[p * kPad + k] = xBase[(size_t)(k0 + k) * xRow + p];
    }
    __syncthreads();
    if (k0 <= m0w + 31) { // causal skip (wave-uniform branch, EXEC stays all-1s)
      mma_panel();
    }
  }

  // ---- Epilogue: out = exp(dA_m)*acc + D[h]*x ----
  const float Dh = Dv[h];
  float* outBase = out + (size_t)((b * kS + c * kCS) * kH + h) * kP;
  const int rAdd = (lane < 16) ? 0 : 8; // C/D layout: VGPR j -> M=j / M=j+8
#pragma unroll
  for (int mt = 0; mt < 2; ++mt) {
#pragma unroll
    for (int nt = 0; nt < 4; ++nt) {
      const int colg = nt * 16 + l16;
#pragma unroll
      for (int j = 0; j < 8; ++j) {
        const int Mg = m0w + mt * 16 + j + rAdd;
        const size_t gi = (size_t)Mg * xRow + colg;
        outBase[gi] = sExp[Mg] * acc[mt][nt][j] + Dh * xBase[gi];
      }
    }
  }
}

extern "C" void kernel_launch(void* const* d_in, const int* /*in_sizes*/, int /*n_in*/,
                              void* d_out, int /*out_size*/, void* /*d_ws*/, size_t /*ws_size*/,
                              hipStream_t stream) {
  const float* cb = (const float*)d_in[0];
  const float* x  = (const float*)d_in[1];
  const float* dt = (const float*)d_in[2];
  const float* dA = (const float*)d_in[3];
  const float* C  = (const float*)d_in[4];
  const float* ps = (const float*)d_in[5];
  const float* D  = (const float*)d_in[6];
  float* out = (float*)d_out;

  const dim3 grid(kB * kNC * kH); // 1024 blocks: one (b, chunk, head) tile each
  const dim3 block(256);          // 8 wave32 waves -> 32 output rows per wave
  hipLaunchKernelGGL(mamba_chunk_scan_wmma_f32, grid, block, 0, stream,
                     cb, x, dt, dA, C, ps, D, out);
}